// APPNPConv_6150393168667
// MI455X (gfx1250) — compile-verified
//
#include <hip/hip_runtime.h>
#include <stdint.h>

// APPNP propagation for MI455X (gfx1250).
// L2-resident scatter/gather SpMM: 10 x (gather * norm -> atomic scatter-add -> axpy).
// CDNA5 paths: global_load_async_to_lds_b128 (ASYNCcnt) stages packed edge records,
// no-return fp32 global atomics (STOREcnt, L2-side RMW) for the scatter, wave32 blocks.

#define DF       48          // feature dim
#define QPE      12          // float4 quads per edge (48/4)
#define EPB      32          // edges per scatter block
#define SCATTER_BLOCK (EPB * QPE)   // 384 threads = 12 waves
#define KSTEPS   10
#define ALPHA    0.1f

__device__ __forceinline__ void atomic_add_f32_noret(float* p, float v) {
  // no-return fp32 atomic, agent scope, relaxed -> global_atomic_add_f32 (no TH_RET)
  (void)__hip_atomic_fetch_add(p, v, __ATOMIC_RELAXED, __HIP_MEMORY_SCOPE_AGENT);
}

// ---- codegen probe: FIRST in file so the disasm snippet shows the atomic lowering.
// Never launched; exists only to verify global_atomic_add_f32 (vs CAS loop) on gfx1250.
__global__ void probe_atomic(float* p, float v) {
  atomic_add_f32_noret(p + threadIdx.x, v);
}

// ---- HOT KERNEL ----
// Block handles 32 edges; thread t -> (edge t/12, float4-quad t%12).
// Wave 0 stages the block's 32 packed 16B edge records into LDS with one
// async b128 per lane; everyone reads them back with one ds_load_b128.
__global__ __launch_bounds__(SCATTER_BLOCK)
void scatter_kernel(const int4* __restrict__ recs, const float* __restrict__ xc,
                    float* __restrict__ agg, int E) {
  __shared__ int4 s_rec[EPB];   // (src, dst, norm_bits, pad) per edge

  const int t  = threadIdx.x;
  const int e0 = blockIdx.x * EPB;

  if (t < EPB) {   // exactly wave 0, full EXEC
    int eg = e0 + t;
    if (eg >= E) eg = E - 1;                       // clamp tail (no OOB)
    unsigned           lds = (unsigned)(unsigned long long)(uintptr_t)&s_rec[t];
    unsigned long long gaddr = (unsigned long long)(uintptr_t)(recs + eg);
    asm volatile("global_load_async_to_lds_b128 %0, %1, off" :: "v"(lds), "v"(gaddr) : "memory");
    asm volatile("s_wait_asynccnt 0" ::: "memory");
  }
  __syncthreads();

  const int el = t / QPE;
  const int q  = t - el * QPE;
  const int e  = e0 + el;
  if (e < E) {
    const int4  r = s_rec[el];
    const int   s = r.x;
    const int   d = r.y;
    const float w = __int_as_float(r.z);
    const float4 v = *(const float4*)(xc + (size_t)s * DF + q * 4);
    float* ap = agg + (size_t)d * DF + q * 4;
    atomic_add_f32_noret(ap + 0, w * v.x);
    atomic_add_f32_noret(ap + 1, w * v.y);
    atomic_add_f32_noret(ap + 2, w * v.z);
    atomic_add_f32_noret(ap + 3, w * v.w);
  }
}

// xn = 0.9*(agg + dinv^2 * xc) + 0.1*h  (self-loop folded analytically), float4-wide.
__global__ __launch_bounds__(256)
void combine_kernel(const float4* __restrict__ agg, const float4* __restrict__ xc,
                    const float4* __restrict__ h, const float* __restrict__ dinv,
                    float4* __restrict__ out, int n4) {
  int gid = blockIdx.x * blockDim.x + threadIdx.x;   // over N*DF/4
  if (gid < n4) {
    int i = gid / QPE;                                // node index
    float di = dinv[i];
    float d2 = di * di;
    float4 a = agg[gid], c = xc[gid], t = h[gid];
    float4 o;
    o.x = (1.0f - ALPHA) * (a.x + d2 * c.x) + ALPHA * t.x;
    o.y = (1.0f - ALPHA) * (a.y + d2 * c.y) + ALPHA * t.y;
    o.z = (1.0f - ALPHA) * (a.z + d2 * c.z) + ALPHA * t.z;
    o.w = (1.0f - ALPHA) * (a.w + d2 * c.w) + ALPHA * t.w;
    out[gid] = o;
  }
}

__global__ __launch_bounds__(256)
void zero4_kernel(float4* __restrict__ p, int n4) {
  int i = blockIdx.x * blockDim.x + threadIdx.x;
  if (i < n4) p[i] = make_float4(0.f, 0.f, 0.f, 0.f);
}

__global__ __launch_bounds__(256)
void deg_init_kernel(float* __restrict__ deg, int n) {
  int i = blockIdx.x * blockDim.x + threadIdx.x;
  if (i < n) deg[i] = 1.0f;                           // self-loop contributes 1
}

__global__ __launch_bounds__(256)
void deg_accum_kernel(const int* __restrict__ dst, float* __restrict__ deg, int E) {
  int e = blockIdx.x * blockDim.x + threadIdx.x;
  if (e < E) atomic_add_f32_noret(&deg[dst[e]], 1.0f);
}

__global__ __launch_bounds__(256)
void dinv_kernel(const float* __restrict__ deg, float* __restrict__ dinv, int n) {
  int i = blockIdx.x * blockDim.x + threadIdx.x;
  if (i < n) {
    float d = deg[i];
    dinv[i] = (d > 0.f) ? rsqrtf(d) : 0.f;
  }
}

// Pack (src, dst, norm, 0) into 16B records: one coalesced b128 per edge thereafter.
__global__ __launch_bounds__(256)
void pack_kernel(const int* __restrict__ src, const int* __restrict__ dst,
                 const float* __restrict__ dinv, int4* __restrict__ recs, int E) {
  int e = blockIdx.x * blockDim.x + threadIdx.x;
  if (e < E) {
    int s = src[e], d = dst[e];
    float w = dinv[s] * dinv[d];
    recs[e] = make_int4(s, d, __float_as_int(w), 0);
  }
}

extern "C" void kernel_launch(void* const* d_in, const int* in_sizes, int n_in,
                              void* d_out, int out_size, void* d_ws, size_t ws_size,
                              hipStream_t stream) {
  const float* x  = (const float*)d_in[0];
  // d_in[1] (x_0) is ignored by the reference module; teleport target is x itself.
  const int*   ei = (const int*)d_in[2];

  const int NF = in_sizes[0];
  const int n  = NF / DF;
  const int E  = in_sizes[2] / 2;
  const int* src  = ei;
  const int* dstp = ei + E;

  // Workspace: deg[n], dinv[n], recs[E] (16B each), bufA[NF], bufB[NF]; 256B aligned.
  char* ws = (char*)d_ws;
  size_t off = 0;
  auto align256 = [](size_t v) { return (v + 255) & ~(size_t)255; };
  float* deg  = (float*)(ws + off); off += align256((size_t)n * 4);
  float* dinv = (float*)(ws + off); off += align256((size_t)n * 4);
  int4*  recs = (int4*) (ws + off); off += align256((size_t)E * 16);
  float* bufA = (float*)(ws + off); off += align256((size_t)NF * 4);
  float* bufB = (float*)(ws + off);

  const int B = 256;

  deg_init_kernel <<<(n + B - 1) / B, B, 0, stream>>>(deg, n);
  deg_accum_kernel<<<(E + B - 1) / B, B, 0, stream>>>(dstp, deg, E);
  dinv_kernel     <<<(n + B - 1) / B, B, 0, stream>>>(deg, dinv, n);
  pack_kernel     <<<(E + B - 1) / B, B, 0, stream>>>(src, dstp, dinv, recs, E);

  float* bufs[2] = {bufA, bufB};
  const int scatterGrid = (E + EPB - 1) / EPB;
  const int n4 = NF / 4;

  for (int k = 0; k < KSTEPS; ++k) {
    const float* xc  = (k == 0) ? x : bufs[(k - 1) & 1];
    float*       agg = bufs[k & 1];
    float*       out = (k == KSTEPS - 1) ? (float*)d_out : agg;   // combine in-place except last

    zero4_kernel   <<<(n4 + B - 1) / B, B, 0, stream>>>((float4*)agg, n4);
    scatter_kernel <<<scatterGrid, SCATTER_BLOCK, 0, stream>>>(recs, xc, agg, E);
    combine_kernel <<<(n4 + B - 1) / B, B, 0, stream>>>((const float4*)agg, (const float4*)xc,
                                                        (const float4*)x, dinv, (float4*)out, n4);
  }
}